// Dfine_70179765616855
// MI455X (gfx1250) — compile-verified
//
#include <hip/hip_runtime.h>
#include <math.h>

// Kalman filter on MI455X (gfx1250).
// B=256 batches, T=256 steps, X=16, U=8, A_DIM=32, fp32 throughout.
// One wave32 per batch; covariance carried in the 8-VGPR WMMA C/D layout;
// all 16x16 block GEMMs on V_WMMA_F32_16X16X4_F32; 32x32 SPD solve via
// wave-parallel Cholesky + triangular solves in LDS.
// Loop-invariant diagonal (Nx / Na) contributions are pre-splatted into
// per-lane v8f registers so the hot loop has no lane-conditional branches
// outside the intrinsically-serial factorization.

#define TSTEPS 256
#define BATCH  256
#define XD 16
#define UD 8
#define AD 32
#define NW 4            // waves (batches) per block

typedef float v2f __attribute__((ext_vector_type(2)));
typedef float v8f __attribute__((ext_vector_type(8)));

__device__ __forceinline__ void wbar() { __builtin_amdgcn_wave_barrier(); }

// ---------------------------------------------------------------------------
// WMMA 16x16x16 block GEMM built from four V_WMMA_F32_16X16X4_F32.
// Operand layouts (CDNA5 ISA 7.12.2):
//   A-op 16x4:  lane l -> M = l&15, holds K = 4*kt + 2*(l>>4) + {0,1}
//   B-op 4x16:  lane l -> N = l&15, holds rows K = 4*kt + 2*(l>>4) + {0,1}
//   C/D 16x16:  vgpr r, lane l -> row = r + 8*(l>>4), col = l&15
// ---------------------------------------------------------------------------
__device__ __forceinline__ v8f mm16_nn(const float* Ap, int lda,
                                       const float* Bp, int ldb,
                                       v8f acc, int lo, int hi) {
#pragma unroll
  for (int kt = 0; kt < 4; ++kt) {
    int k = 4 * kt + 2 * hi;
    v2f a = { Ap[lo * lda + k], Ap[lo * lda + k + 1] };
    v2f b = { Bp[k * ldb + lo], Bp[(k + 1) * ldb + lo] };
    acc = __builtin_amdgcn_wmma_f32_16x16x4_f32(false, a, false, b,
                                                (short)0, acc, false, false);
  }
  return acc;
}

// B operand is (row-major Bm) transposed:  B_op[k][n] = Bm[n][k]
__device__ __forceinline__ v8f mm16_nt(const float* Ap, int lda,
                                       const float* Bm, int ldb,
                                       v8f acc, int lo, int hi) {
#pragma unroll
  for (int kt = 0; kt < 4; ++kt) {
    int k = 4 * kt + 2 * hi;
    v2f a = { Ap[lo * lda + k], Ap[lo * lda + k + 1] };
    v2f b = { Bm[lo * ldb + k], Bm[lo * ldb + k + 1] };
    acc = __builtin_amdgcn_wmma_f32_16x16x4_f32(false, a, false, b,
                                                (short)0, acc, false, false);
  }
  return acc;
}

// A operand is (row-major Am) transposed:  A_op[m][k] = Am[k][m]
__device__ __forceinline__ v8f mm16_tn(const float* Am, int lda,
                                       const float* Bp, int ldb,
                                       v8f acc, int lo, int hi) {
#pragma unroll
  for (int kt = 0; kt < 4; ++kt) {
    int k = 4 * kt + 2 * hi;
    v2f a = { Am[k * lda + lo], Am[(k + 1) * lda + lo] };
    v2f b = { Bp[k * ldb + lo], Bp[(k + 1) * ldb + lo] };
    acc = __builtin_amdgcn_wmma_f32_16x16x4_f32(false, a, false, b,
                                                (short)0, acc, false, false);
  }
  return acc;
}

__device__ __forceinline__ float softplusf(float x) {
  return (x > 20.f) ? x : log1pf(expf(x));
}

// ---------------------------------------------------------------------------
// Prep kernel: build A = Uq diag(sqrt(sp(d))) Q diag(1/sqrt(1+sp(d))) Uq^T
// via modified Gram-Schmidt QR (positive R diag == reference's sign fix),
// plus C^T and softplus'ed noise diagonals into workspace.
// ws layout (floats): [0,256) A rowmajor | [256,768) Ct 16x32 | [768,784) nxp
//                     [784,816) nap
// ---------------------------------------------------------------------------
__device__ void mgs16(const float* Min, float Q[16][16]) {
  for (int i = 0; i < 16; ++i)
    for (int j = 0; j < 16; ++j) Q[i][j] = Min[i * 16 + j];
  for (int j = 0; j < 16; ++j) {
    for (int p = 0; p < j; ++p) {
      float r = 0.f;
      for (int k = 0; k < 16; ++k) r += Q[k][p] * Q[k][j];
      for (int k = 0; k < 16; ++k) Q[k][j] -= r * Q[k][p];
    }
    float nrm = 0.f;
    for (int k = 0; k < 16; ++k) nrm += Q[k][j] * Q[k][j];
    float inv = rsqrtf(fmaxf(nrm, 1e-30f));
    for (int k = 0; k < 16; ++k) Q[k][j] *= inv;
  }
}

__global__ void kf_prep(const float* __restrict__ M, const float* __restrict__ N,
                        const float* __restrict__ d, const float* __restrict__ C,
                        const float* __restrict__ nx, const float* __restrict__ na,
                        float* __restrict__ ws) {
  int tid = threadIdx.x;
  for (int i = tid; i < AD * XD; i += blockDim.x) {
    int n = i / XD, k = i % XD;
    ws[256 + k * AD + n] = C[n * XD + k];          // Ct[k][n] = C[n][k]
  }
  for (int i = tid; i < XD; i += blockDim.x)
    ws[768 + i] = softplusf(nx[i]) + 1e-4f;
  for (int i = tid; i < AD; i += blockDim.x)
    ws[784 + i] = softplusf(na[i]) + 1e-4f;

  if (tid == 0) {
    float Qm[16][16], Uq[16][16], T[16][16], tmp[16][16];
    mgs16(M, Qm);
    mgs16(N, Uq);
    float sd[16], si[16];
    for (int i = 0; i < 16; ++i) {
      float sp = softplusf(d[i]);
      sd[i] = sqrtf(sp);
      si[i] = rsqrtf(1.f + sp);
    }
    for (int i = 0; i < 16; ++i)
      for (int j = 0; j < 16; ++j) T[i][j] = sd[i] * Qm[i][j] * si[j];
    for (int i = 0; i < 16; ++i)
      for (int j = 0; j < 16; ++j) {
        float s = 0.f;
        for (int k = 0; k < 16; ++k) s += Uq[i][k] * T[k][j];
        tmp[i][j] = s;
      }
    for (int i = 0; i < 16; ++i)
      for (int j = 0; j < 16; ++j) {
        float s = 0.f;
        for (int k = 0; k < 16; ++k) s += tmp[i][k] * Uq[j][k];  // * Uq^T
        ws[i * 16 + j] = s;
      }
  }
}

// ---------------------------------------------------------------------------
// Main kernel: 4 waves / block, one batch per wave. Covariance lives in a v8f
// (WMMA C/D layout); LDS used for operand staging, Cholesky and solves.
// ---------------------------------------------------------------------------
__global__ __launch_bounds__(32 * NW, 1) void kf_scan(
    const float* __restrict__ mean0, const float* __restrict__ cov0,
    const float* __restrict__ u, const float* __restrict__ a,
    const float* __restrict__ Bm, const float* __restrict__ C,
    const float* __restrict__ ws, float* __restrict__ out) {
  __shared__ float sA[256];        // A row-major
  __shared__ float sCt[512];       // C^T 16x32
  __shared__ float sC[512];        // C 32x16
  __shared__ float sB[128];        // B 16x8
  __shared__ float sNx[16];
  __shared__ float sNa[32];
  __shared__ float sMtx[NW][256];      // 16x16 staging
  __shared__ float sP[NW][512];        // P = C*cov_pred, 32x16
  __shared__ float sR[NW][32 * 18];    // RHS -> Y -> W (cols 0..15) + innov (col 16)
  __shared__ float sS[NW][32 * 33];    // S / Cholesky factor, padded rows
  __shared__ float sMean[NW][16];
  __shared__ float sU[NW][8];

  int tid = threadIdx.x;
  for (int i = tid; i < 256; i += blockDim.x) sA[i] = ws[i];
  for (int i = tid; i < 512; i += blockDim.x) sCt[i] = ws[256 + i];
  for (int i = tid; i < 512; i += blockDim.x) sC[i] = C[i];
  for (int i = tid; i < 128; i += blockDim.x) sB[i] = Bm[i];
  for (int i = tid; i < 16; i += blockDim.x) sNx[i] = ws[768 + i];
  for (int i = tid; i < 32; i += blockDim.x) sNa[i] = ws[784 + i];
  __syncthreads();

  const int w = tid >> 5, lane = tid & 31;
  const int lo = lane & 15, hi = lane >> 4;
  const int b = blockIdx.x * NW + w;

  float* Mw = sMtx[w];
  float* Pw = sP[w];
  float* Rw = sR[w];
  float* Sw = sS[w];
  float* mv = sMean[w];
  float* uv = sU[w];

  const v8f zero = {0.f, 0.f, 0.f, 0.f, 0.f, 0.f, 0.f, 0.f};

  // Loop-invariant per-lane diagonal splats (no branches in the hot loop):
  // element r of a C/D fragment is (row = r+8*hi, col = lo).
  v8f nxdiag, nadiag0, nadiag1;
#pragma unroll
  for (int r = 0; r < 8; ++r) {
    const bool diag = (lo == r + 8 * hi);
    nxdiag[r]  = diag ? sNx[lo]      : 0.f;
    nadiag0[r] = diag ? sNa[lo]      : 0.f;
    nadiag1[r] = diag ? sNa[16 + lo] : 0.f;
  }

  v8f covr;
#pragma unroll
  for (int r = 0; r < 8; ++r)
    covr[r] = cov0[b * 256 + (r + 8 * hi) * 16 + lo];
  if (lane < 16) mv[lo] = mean0[b * 16 + lo];
  wbar();

  for (int t = 0; t < TSTEPS; ++t) {
    // ---- dynamics: mean = A*mean + B*u_t -------------------------------
    if (lane < 8) uv[lane] = u[(b * TSTEPS + t) * UD + lane];
    wbar();
    float mn = 0.f;
    if (lane < 16) {
      for (int j = 0; j < 16; ++j) mn = fmaf(sA[lo * 16 + j], mv[j], mn);
      for (int k = 0; k < 8; ++k) mn = fmaf(sB[lo * 8 + k], uv[k], mn);
    }
    wbar();
    if (lane < 16) mv[lo] = mn;
    wbar();
    // innovation = a_t - C*mean  (stored as 17th RHS column)
    float inn = a[(size_t)(b * TSTEPS + t) * AD + lane];
    for (int m = 0; m < 16; ++m) inn = fmaf(-sC[lane * 16 + m], mv[m], inn);
    Rw[lane * 18 + 16] = inn;

    // ---- cov_pred = A cov A^T + Nx  (WMMA) ------------------------------
#pragma unroll
    for (int r = 0; r < 8; ++r) Mw[(r + 8 * hi) * 16 + lo] = covr[r];
    wbar();
    v8f t1 = mm16_nn(sA, 16, Mw, 16, zero, lo, hi);      // A * cov
    wbar();
#pragma unroll
    for (int r = 0; r < 8; ++r) Mw[(r + 8 * hi) * 16 + lo] = t1[r];
    wbar();
    v8f covp = mm16_nt(Mw, 16, sA, 16, zero, lo, hi);    // (A cov) * A^T
    covp += nxdiag;                                      // + Nx (branchless)
#pragma unroll
    for (int r = 0; r < 8; ++r) Mw[(r + 8 * hi) * 16 + lo] = covp[r];
    wbar();

    // ---- P = C * cov_pred (32x16), also RHS cols 0..15 ------------------
#pragma unroll
    for (int i = 0; i < 2; ++i) {
      v8f p = mm16_nn(sC + i * 256, 16, Mw, 16, zero, lo, hi);
#pragma unroll
      for (int r = 0; r < 8; ++r) {
        int row = 16 * i + r + 8 * hi;
        Pw[row * 16 + lo] = p[r];
        Rw[row * 18 + lo] = p[r];
      }
    }
    wbar();

    // ---- S = P * C^T + diag(Na)  (4 WMMA blocks) ------------------------
#pragma unroll
    for (int i = 0; i < 2; ++i)
#pragma unroll
      for (int j = 0; j < 2; ++j) {
        v8f s = mm16_nn(Pw + i * 256, 16, sCt + 16 * j, 32, zero, lo, hi);
        if (i == 0 && j == 0) s += nadiag0;   // compile-time conditions,
        if (i == 1 && j == 1) s += nadiag1;   // branchless diagonal adds
#pragma unroll
        for (int r = 0; r < 8; ++r)
          Sw[(16 * i + r + 8 * hi) * 33 + 16 * j + lo] = s[r];
      }
    wbar();

    // ---- Cholesky: S = L L^T (right-looking, wave-parallel) -------------
    for (int k = 0; k < 32; ++k) {
      float dkk = Sw[k * 33 + k];
      float inv = rsqrtf(fmaxf(dkk, 1e-30f));
      float lik = 0.f;
      if (lane >= k) { lik = Sw[lane * 33 + k] * inv; Sw[lane * 33 + k] = lik; }
      wbar();
      if (lane > k) {
        for (int tt = k + 1; tt <= lane; ++tt)
          Sw[lane * 33 + tt] = fmaf(-lik, Sw[tt * 33 + k], Sw[lane * 33 + tt]);
      }
      wbar();
    }

    // ---- solves: L Y = [P|innov], then L^T W = Y  (17 RHS cols) ---------
    if (lane < 17) {
      for (int r = 0; r < 32; ++r) {
        float acc = Rw[r * 18 + lane];
        for (int tt = 0; tt < r; ++tt)
          acc = fmaf(-Sw[r * 33 + tt], Rw[tt * 18 + lane], acc);
        Rw[r * 18 + lane] = acc / Sw[r * 33 + r];
      }
      for (int r = 31; r >= 0; --r) {
        float acc = Rw[r * 18 + lane];
        for (int tt = r + 1; tt < 32; ++tt)
          acc = fmaf(-Sw[tt * 33 + r], Rw[tt * 18 + lane], acc);
        Rw[r * 18 + lane] = acc / Sw[r * 33 + r];
      }
    }
    wbar();

    // ---- cov = cov_pred - P^T W   (WMMA, subtract in VALU: f32 has no
    //      A-negate, only C-negate per ISA NEG table) ----------------------
    v8f kc = zero;
#pragma unroll
    for (int i = 0; i < 2; ++i)
      kc = mm16_tn(Pw + i * 256, 16, Rw + i * 16 * 18, 18, kc, lo, hi);
    covr = covp - kc;

    // ---- mean = mean + P^T (S^-1 innov); emit filtered mean -------------
    if (lane < 16) {
      float mfin = mv[lo];
      for (int r = 0; r < 32; ++r)
        mfin = fmaf(Pw[r * 16 + lo], Rw[r * 18 + 16], mfin);
      out[(size_t)(b * TSTEPS + t) * XD + lo] = mfin;
      mv[lo] = mfin;
    }
    wbar();
  }
}

extern "C" void kernel_launch(void* const* d_in, const int* in_sizes, int n_in,
                              void* d_out, int out_size, void* d_ws, size_t ws_size,
                              hipStream_t stream) {
  const float* mean0 = (const float*)d_in[0];
  const float* cov0  = (const float*)d_in[1];
  const float* u     = (const float*)d_in[2];
  const float* a     = (const float*)d_in[3];
  const float* M     = (const float*)d_in[4];
  const float* N     = (const float*)d_in[5];
  const float* dvec  = (const float*)d_in[6];
  const float* B     = (const float*)d_in[7];
  const float* C     = (const float*)d_in[8];
  const float* nx    = (const float*)d_in[9];
  const float* na    = (const float*)d_in[10];
  float* ws  = (float*)d_ws;
  float* out = (float*)d_out;

  kf_prep<<<1, 64, 0, stream>>>(M, N, dvec, C, nx, na, ws);
  kf_scan<<<BATCH / NW, 32 * NW, 0, stream>>>(mean0, cov0, u, a, B, C, ws, out);
}